// HeteroHyperConvLayer_20358144983738
// MI455X (gfx1250) — compile-verified
//
#include <hip/hip_runtime.h>

// -----------------------------------------------------------------------------
// HeteroHyperConv for MI455X (gfx1250, wave32).
//
// Roofline: dense GEMMs total ~8 GFLOP (trivial). The two SpMMs move ~3.3 GB of
// gathered rows + ~820M f32 atomic adds; poi_proj (51MB) and fused_edge (26MB)
// fit in the 192MB L2, so gathers/atomics are L2-resident. Memory-bound ->
// keep GEMMs in full f32 via V_WMMA_F32_16X16X4_F32 (exact-precision path),
// and make the SpMM a wave-per-nnz float4 gather + global_atomic_add_f32.
// -----------------------------------------------------------------------------

typedef __attribute__((ext_vector_type(2))) float v2f;
typedef __attribute__((ext_vector_type(8))) float v8f;

#define EMB 128

__device__ __forceinline__ v8f wmma_f32_16x16x4(v2f a, v2f b, v8f c) {
    // 8 args: (neg_a, A, neg_b, B, c_mod, C, reuse_a, reuse_b)
    return __builtin_amdgcn_wmma_f32_16x16x4_f32(
        false, a, false, b, (short)0, c, false, false);
}

// C[M,128] = A[M,128] @ W[128,128], f32, one 16x16 WMMA tile per wave.
// Block = 256 threads = 8 waves; wave w owns N-tile [16w,16w+16); block b owns
// M rows [16b,16b+16). K=128 -> 32 chained V_WMMA_F32_16X16X4_F32.
__global__ void __launch_bounds__(256)
gemm128_wmma(const float* __restrict__ A, const float* __restrict__ W,
             float* __restrict__ C)
{
    const int lane = threadIdx.x & 31;
    const int wave = threadIdx.x >> 5;
    const int m0   = blockIdx.x << 4;
    const int n0   = wave << 4;
    const int half = lane >> 4;     // 0: K pair {0,1}; 1: K pair {2,3}
    const int l    = lane & 15;

    const float* arow = A + (size_t)(m0 + l) * EMB;  // A: M = lane&15
    v8f acc = {};
#pragma unroll
    for (int k0 = 0; k0 < EMB; k0 += 4) {
        const int ka = k0 + (half << 1);
        // A 16x4 frag: vgpr0 = K=0 (lanes 0-15) / K=2 (lanes 16-31); vgpr1 = K=1/K=3
        v2f a = *(const v2f*)(arow + ka);            // 8B-aligned (row is 512B aligned)
        // B 4x16 frag: mirrored layout over N = lane&15
        v2f b;
        b.x = W[(size_t)ka * EMB + n0 + l];
        b.y = W[(size_t)(ka + 1) * EMB + n0 + l];
        acc = wmma_f32_16x16x4(a, b, acc);
    }
    // C/D layout: vgpr v -> row m0 + v + 8*half, col n0 + (lane&15)
    float* crow = C + (size_t)(m0 + half * 8) * EMB + n0 + l;
#pragma unroll
    for (int v = 0; v < 8; ++v) crow[(size_t)v * EMB] = acc[v];
}

// C[M,128] = A1[M,128] @ W[0:128,:] + A2[M,128] @ W[128:256,:]
// (fused concat-GEMM: one accumulator over K=256, two A sources)
__global__ void __launch_bounds__(256)
gemm_fused_wmma(const float* __restrict__ A1, const float* __restrict__ A2,
                const float* __restrict__ W, float* __restrict__ C)
{
    const int lane = threadIdx.x & 31;
    const int wave = threadIdx.x >> 5;
    const int m0   = blockIdx.x << 4;
    const int n0   = wave << 4;
    const int half = lane >> 4;
    const int l    = lane & 15;

    const float* a1row = A1 + (size_t)(m0 + l) * EMB;
    const float* a2row = A2 + (size_t)(m0 + l) * EMB;
    v8f acc = {};
#pragma unroll
    for (int k0 = 0; k0 < EMB; k0 += 4) {
        const int ka = k0 + (half << 1);
        v2f a = *(const v2f*)(a1row + ka);
        v2f b;
        b.x = W[(size_t)ka * EMB + n0 + l];
        b.y = W[(size_t)(ka + 1) * EMB + n0 + l];
        acc = wmma_f32_16x16x4(a, b, acc);
    }
#pragma unroll
    for (int k0 = 0; k0 < EMB; k0 += 4) {
        const int ka = k0 + (half << 1);
        v2f a = *(const v2f*)(a2row + ka);
        v2f b;
        b.x = W[(size_t)(EMB + ka) * EMB + n0 + l];
        b.y = W[(size_t)(EMB + ka + 1) * EMB + n0 + l];
        acc = wmma_f32_16x16x4(a, b, acc);
    }
    float* crow = C + (size_t)(m0 + half * 8) * EMB + n0 + l;
#pragma unroll
    for (int v = 0; v < 8; ++v) crow[(size_t)v * EMB] = acc[v];
}

// COO SpMM scatter: out[rows[i], :] += vals[i] * dense[cols[i], :]
// One wave per nnz: each lane gathers a float4 (global_load_b128, fully
// coalesced 512B row) and issues 4 global_atomic_add_f32 (resolved in L2).
__global__ void __launch_bounds__(256)
spmm_atomic(const int* __restrict__ rows, const int* __restrict__ cols,
            const float* __restrict__ vals, const float* __restrict__ dense,
            float* __restrict__ out, int nnz)
{
    const int  lane   = threadIdx.x & 31;
    const int  wv     = threadIdx.x >> 5;
    long long  i      = (long long)blockIdx.x * 8 + wv;
    const long long s = (long long)gridDim.x * 8;

    for (; i < nnz; i += s) {
        const int   r = __builtin_amdgcn_readfirstlane(rows[i]);
        const int   c = __builtin_amdgcn_readfirstlane(cols[i]);
        const float v = vals[i];  // wave-uniform

        const float4 d = ((const float4*)(dense + (size_t)c * EMB))[lane];
        float* dst = out + (size_t)r * EMB + (lane << 2);
        __hip_atomic_fetch_add(dst + 0, v * d.x, __ATOMIC_RELAXED, __HIP_MEMORY_SCOPE_AGENT);
        __hip_atomic_fetch_add(dst + 1, v * d.y, __ATOMIC_RELAXED, __HIP_MEMORY_SCOPE_AGENT);
        __hip_atomic_fetch_add(dst + 2, v * d.z, __ATOMIC_RELAXED, __HIP_MEMORY_SCOPE_AGENT);
        __hip_atomic_fetch_add(dst + 3, v * d.w, __ATOMIC_RELAXED, __HIP_MEMORY_SCOPE_AGENT);
    }
}

__global__ void __launch_bounds__(256)
zero_f32(float4* __restrict__ p, long long n4)
{
    long long i = (long long)blockIdx.x * blockDim.x + threadIdx.x;
    const long long s = (long long)gridDim.x * blockDim.x;
    for (; i < n4; i += s) p[i] = make_float4(0.f, 0.f, 0.f, 0.f);
}

extern "C" void kernel_launch(void* const* d_in, const int* in_sizes, int n_in,
                              void* d_out, int out_size, void* d_ws, size_t ws_size,
                              hipStream_t stream)
{
    const float* poi_embs  = (const float*)d_in[0];
    const float* edge_embs = (const float*)d_in[1];
    const int*   pte_rows  = (const int*)d_in[2];
    const int*   pte_cols  = (const int*)d_in[3];
    const float* pte_vals  = (const float*)d_in[4];
    const int*   etp_rows  = (const int*)d_in[5];
    const int*   etp_cols  = (const int*)d_in[6];
    const float* etp_vals  = (const float*)d_in[7];
    const float* W_poi     = (const float*)d_in[8];
    const float* W_edge    = (const float*)d_in[9];
    const float* W_fusion  = (const float*)d_in[10];

    const int n_poi  = in_sizes[0] / EMB;   // 100000 (multiple of 16)
    const int n_edge = in_sizes[1] / EMB;   // 50000  (multiple of 16)
    const int nnz    = in_sizes[2];         // 3.2M

    float* out        = (float*)d_out;
    float* propagated = out;                            // [n_poi, 128]
    float* fused      = out + (size_t)n_poi * EMB;      // [n_edge, 128]

    float* poi_proj = (float*)d_ws;                     // [n_poi, 128]  51.2 MB
    float* poi_msg  = poi_proj + (size_t)n_poi * EMB;   // [n_edge, 128] 25.6 MB
    float* edge_msg = poi_msg + (size_t)n_edge * EMB;   // [n_edge, 128] 25.6 MB

    const int spmm_blocks = (nnz + 7) / 8;              // 8 waves / block

    // 1) poi_proj = poi_embs @ W_poi
    gemm128_wmma<<<n_poi / 16, 256, 0, stream>>>(poi_embs, W_poi, poi_proj);

    // 2) poi_msg = spmm(pte, poi_proj)  [segment_sum -> zero + atomic scatter]
    {
        const long long n4 = (long long)n_edge * EMB / 4;
        zero_f32<<<(int)((n4 + 255) / 256), 256, 0, stream>>>((float4*)poi_msg, n4);
    }
    spmm_atomic<<<spmm_blocks, 256, 0, stream>>>(pte_rows, pte_cols, pte_vals,
                                                 poi_proj, poi_msg, nnz);

    // 3) edge_msg = edge_embs @ W_edge
    gemm128_wmma<<<n_edge / 16, 256, 0, stream>>>(edge_embs, W_edge, edge_msg);

    // 4) fused_edge = [poi_msg | edge_msg] @ W_fusion  (fused, no concat buffer)
    gemm_fused_wmma<<<n_edge / 16, 256, 0, stream>>>(poi_msg, edge_msg, W_fusion, fused);

    // 5) propagated_poi = spmm(etp, fused_edge)
    {
        const long long n4 = (long long)n_poi * EMB / 4;
        zero_f32<<<(int)((n4 + 255) / 256), 256, 0, stream>>>((float4*)propagated, n4);
    }
    spmm_atomic<<<spmm_blocks, 256, 0, stream>>>(etp_rows, etp_cols, etp_vals,
                                                 fused, propagated, nnz);
}